// AttentionMono_67199058313423
// MI455X (gfx1250) — compile-verified
//
#include <hip/hip_runtime.h>
#include <hip/hip_bf16.h>

// ---------------------------------------------------------------------------
// AttentionMono on MI455X (gfx1250, wave32, WMMA + TDM)
//
// B=8, K=Q=T=256, A=64.
// Fused per-(b,k) formulation avoids the 134MB [B,K,Q,A] intermediate:
//   logits[c][q] = sum_a (Wa[c,a]*key[b,k,a]) * query[b,q,a]   (64x256, K=64)
// computed entirely in LDS with V_WMMA_F32_16X16X4_F32. The 64KB query tile is
// staged into LDS by the Tensor Data Mover (tensor_load_to_lds) with hardware
// padding that reproduces the bank-conflict-free stride-68 row layout.
// ---------------------------------------------------------------------------

typedef float v2f __attribute__((ext_vector_type(2)));
typedef float v4f __attribute__((ext_vector_type(4)));
typedef float v8f __attribute__((ext_vector_type(8)));
typedef unsigned int u32x4 __attribute__((ext_vector_type(4)));
typedef int i32x4 __attribute__((ext_vector_type(4)));
typedef int i32x8 __attribute__((ext_vector_type(8)));

__device__ __forceinline__ v8f wmma4(v2f a, v2f b, v8f c) {
  // D = A(16x4) * B(4x16) + C(16x16), all f32.
  // 8-arg form: (neg_a, A, neg_b, B, c_mod, C, reuse_a, reuse_b)
  return __builtin_amdgcn_wmma_f32_16x16x4_f32(false, a, false, b, (short)0, c,
                                               false, false);
}

#define NB   8
#define NK   256
#define NQ   256
#define NT   256
#define NA   64

#define QSTR 68    // query LDS row stride (floats): conflict-free b64 frags
#define WSTR 68    // KW LDS row stride
#define LSTR 257   // logits LDS row stride: conflict-free row-wise reduction

// ---------------------------------------------------------------------------
// Kernel 1: key/query/value projections.
//   X [2048, 256] (row-major) x W^T (W is [64,256] row-major) -> O [2048, 64]
// grid = (64 row-groups of 32 rows, 3 matrices), block = 256 (8 waves).
// ---------------------------------------------------------------------------
__global__ __launch_bounds__(256) void proj_kernel(
    const float* __restrict__ keyT, const float* __restrict__ qryT,
    const float* __restrict__ Wk,   const float* __restrict__ Wq,
    const float* __restrict__ Wvd,
    float* __restrict__ keyO, float* __restrict__ qryO, float* __restrict__ valO) {
  const int wave = threadIdx.x >> 5;
  const int lane = threadIdx.x & 31;
  const int hl   = lane >> 4;       // half-wave: k offset 0 / 2
  const int l16  = lane & 15;

  const float* X; const float* W; float* O;
  if (blockIdx.y == 0)      { X = keyT; W = Wk;  O = keyO; }
  else if (blockIdx.y == 1) { X = qryT; W = Wq;  O = qryO; }
  else                      { X = keyT; W = Wvd; O = valO; }

  const int rowTile = blockIdx.x * 2 + (wave >> 2); // 2 row tiles per WG
  const int nTile   = wave & 3;                     // A=64 -> 4 col tiles
  const int m = rowTile * 16 + l16;                 // A-frag row
  const int n = nTile * 16 + l16;                   // B-frag col (= W row)

  v8f acc = {};
  for (int kk = 0; kk < NT; kk += 4) {
    const int k0 = kk + 2 * hl;
    v2f af = *(const v2f*)(X + m * NT + k0);        // A[m][k0..k0+1]
    v2f bf = *(const v2f*)(W + n * NT + k0);        // B[k][n] = W[n][k]
    acc = wmma4(af, bf, acc);
  }
#pragma unroll
  for (int i = 0; i < 8; ++i) {
    const int mm = rowTile * 16 + i + 8 * hl;       // D layout: M = i + 8*half
    O[mm * NA + nTile * 16 + l16] = acc[i];
  }
}

// ---------------------------------------------------------------------------
// Kernel 2: per-(b,k) fused logits GEMM + swishmax + K-sum accumulation.
// grid = B*K = 2048 workgroups, block = 256 (8 waves), ~151 KB dynamic LDS.
// ---------------------------------------------------------------------------
__global__ __launch_bounds__(256) void attn_kernel(
    const float* __restrict__ key, const float* __restrict__ qry,
    const float* __restrict__ val, const float* __restrict__ Wa,
    float* __restrict__ vsum) {
  extern __shared__ __align__(16) float smem[];
  float* qsh  = smem;                    // [256][QSTR]  query[b], padded
  float* kwsh = qsh  + NQ * QSTR;        // [64][WSTR]   KW = Wa o key
  float* lg   = kwsh + NA * WSTR;        // [64][LSTR]   logits -> xexp
  float* keyv = lg   + NA * LSTR;        // [64]
  float* wv   = keyv + NA;               // [64]  value / denom
  float* red  = wv   + NA;               // [256] reduction scratch

  const int tid = threadIdx.x;
  const int b = blockIdx.x >> 8;
  const int k = blockIdx.x & 255;

#if __has_builtin(__builtin_amdgcn_tensor_load_to_lds) && \
    __has_builtin(__builtin_amdgcn_s_wait_tensorcnt)
  // -------- TDM path: one descriptor copies query[b] (256x64 f32) into LDS
  // with 4-DWORD padding after every 64 DWORDs => row stride 68 floats.
  if (tid < 32) {                                   // wave 0 issues the DMA
    const unsigned lds_base =
        (unsigned)(unsigned long long)(const void*)qsh;     // LDS byte offset
    const unsigned long long ga =
        (unsigned long long)(const void*)(qry + (size_t)b * NQ * NA);
    // D# group0: count=1 | lds_addr | global_addr[56:0] | type=2
    u32x4 g0 = { 1u, lds_base, (unsigned)ga,
                 (unsigned)((ga >> 32) & 0x01FFFFFFu) | 0x80000000u };
    // D# group1: data_size=4B, pad_enable, pad_interval=64dw(5), pad_amount=4dw(3)
    //            tensor_dim0=64, tensor_dim1=256, tile_dim0=64, tile_dim1=256,
    //            tensor_dim0_stride=64
    i32x8 g1 = { (int)0x07520000,
                 (int)(64u << 16),    // tensor_dim0[15:0] @ bit48
                 (int)(256u << 16),   // tensor_dim1[15:0] @ bit80
                 (int)(64u << 16),    // tile_dim0 @ bit112
                 (int)256,            // tile_dim1 @ bit128
                 (int)64,             // tensor_dim0_stride[31:0] @ bit160
                 0, 0 };
    i32x4 gz4 = { 0, 0, 0, 0 };       // groups 2/3 unused (tile_dim2..4 = 0)
    i32x8 gz8 = { 0, 0, 0, 0, 0, 0, 0, 0 };
    // 6-arg toolchain form: (g0, g1, g2, g3, extra_group, cpol)
    __builtin_amdgcn_tensor_load_to_lds(g0, g1, gz4, gz4, gz8, 0);
  }
  if (tid >= 64 && tid < 128)  keyv[tid - 64]  = key[(b * NK + k) * NA + (tid - 64)];
  if (tid >= 128 && tid < 192) wv[tid - 128]   = val[(b * NK + k) * NA + (tid - 128)];
  if (tid < 32) __builtin_amdgcn_s_wait_tensorcnt(0);
#else
  // -------- fallback: vectorized manual staging
  if (tid >= 64 && tid < 128)  keyv[tid - 64]  = key[(b * NK + k) * NA + (tid - 64)];
  if (tid >= 128 && tid < 192) wv[tid - 128]   = val[(b * NK + k) * NA + (tid - 128)];
  for (int i = tid * 4; i < NQ * NA; i += 256 * 4) {
    const int q = i >> 6, a = i & 63;               // rows of 64: a % 4 == 0
    *(v4f*)(qsh + q * QSTR + a) = *(const v4f*)(qry + (size_t)b * NQ * NA + i);
  }
#endif
  __syncthreads();

  for (int i = tid; i < NA * NA; i += 256) {        // KW[c][a] = Wa[c,a]*key[a]
    const int c = i >> 6, a = i & 63;
    kwsh[c * WSTR + a] = Wa[i] * keyv[a];
  }
  __syncthreads();

  // logits[c][q] = sum_a KW[c][a] * query[q][a]  (M=64, N=256, Kdim=64)
  const int wave = tid >> 5, lane = tid & 31, hl = lane >> 4, l16 = lane & 15;
  for (int t = wave; t < 64; t += 8) {              // 4 cTiles x 16 qTiles
    const int cT = t >> 4, qT = t & 15;
    v8f acc = {};
#pragma unroll
    for (int kk = 0; kk < NA; kk += 4) {
      const int k0 = kk + 2 * hl;
      v2f af = *(const v2f*)(kwsh + (cT * 16 + l16) * WSTR + k0);
      v2f bf = *(const v2f*)(qsh  + (qT * 16 + l16) * QSTR + k0);
      acc = wmma4(af, bf, acc);
    }
#pragma unroll
    for (int i = 0; i < 8; ++i)
      lg[(cT * 16 + i + 8 * hl) * LSTR + qT * 16 + l16] = acc[i];
  }
  __syncthreads();

  // swishmax over q per row c: 4 threads per row (c = tid&63, part = tid>>6)
  const int c = tid & 63, p = tid >> 6;
  float mx = -3.402823466e38f;
  for (int q = p * 64; q < p * 64 + 64; ++q) mx = fmaxf(mx, lg[c * LSTR + q]);
  red[tid] = mx;
  __syncthreads();
  const float rowmax = fmaxf(fmaxf(red[c], red[64 + c]),
                             fmaxf(red[128 + c], red[192 + c]));
  float s = 0.f;
  for (int q = p * 64; q < p * 64 + 64; ++q) {      // xexp in place + |.|-sum
    const float x  = lg[c * LSTR + q];
    const float xe = x * __expf(x - rowmax);
    lg[c * LSTR + q] = xe;
    s += fabsf(xe);
  }
  __syncthreads();                                  // all rowmax reads done
  red[tid] = s;
  __syncthreads();
  if (tid < 64) {                                   // fold 1/denom into value
    const float denom = red[c] + red[64 + c] + red[128 + c] + red[192 + c] + 1.0f;
    wv[c] = wv[c] / denom;
  }
  __syncthreads();

  // vsum[b,q,a] += value[a]/denom[a] * xexp[a][q]   (thread tid owns column q)
  float* dst = vsum + (b * NQ + tid) * NA;
  for (int a = 0; a < NA; ++a)
    atomicAdd(dst + a, wv[a] * lg[a * LSTR + tid]);
}

// ---------------------------------------------------------------------------
// Kernel 3: out[b,q,t] = sum_a vsum[b,q,a] * Wvu[t,a]
//   [2048, 64] x [64, 256] -> [2048, 256]; one 16x16 tile per wave.
// ---------------------------------------------------------------------------
__global__ __launch_bounds__(256) void out_kernel(
    const float* __restrict__ vsum, const float* __restrict__ Wvu,
    float* __restrict__ out) {
  const int wave = threadIdx.x >> 5, lane = threadIdx.x & 31;
  const int hl = lane >> 4, l16 = lane & 15;
  const int tile = blockIdx.x * 8 + wave;           // 0..2047
  const int rT = tile >> 4;                         // 128 row tiles
  const int tT = tile & 15;                         // 16 col tiles
  const int m = rT * 16 + l16;
  const int n = tT * 16 + l16;

  v8f acc = {};
#pragma unroll
  for (int kk = 0; kk < NA; kk += 4) {
    const int k0 = kk + 2 * hl;
    v2f af = *(const v2f*)(vsum + m * NA + k0);
    v2f bf = *(const v2f*)(Wvu + n * NA + k0);      // B[k][n] = Wvu[n][k]
    acc = wmma4(af, bf, acc);
  }
#pragma unroll
  for (int i = 0; i < 8; ++i)
    out[(rT * 16 + i + 8 * hl) * NT + tT * 16 + l16] = acc[i];
}

// ---------------------------------------------------------------------------
extern "C" void kernel_launch(void* const* d_in, const int* in_sizes, int n_in,
                              void* d_out, int out_size, void* d_ws, size_t ws_size,
                              hipStream_t stream) {
  const float* key_tokens   = (const float*)d_in[0]; // [8,256,256]
  const float* query_tokens = (const float*)d_in[1]; // [8,256,256]
  const float* Wk  = (const float*)d_in[2];          // [64,256]
  const float* Wq  = (const float*)d_in[3];          // [64,256]
  const float* Wa  = (const float*)d_in[4];          // [64,64]
  const float* Wvd = (const float*)d_in[5];          // [64,256]
  const float* Wvu = (const float*)d_in[6];          // [256,64]
  float* out = (float*)d_out;                        // [8,256,256]

  float* key  = (float*)d_ws;                        // [2048,64]
  float* qry  = key  + NB * NK * NA;
  float* val  = qry  + NB * NQ * NA;
  float* vsum = val  + NB * NK * NA;                 // [2048,64] accumulator

  // graph-capture-safe async memset of the K-sum accumulator
  (void)hipMemsetAsync(vsum, 0, (size_t)NB * NQ * NA * sizeof(float), stream);

  proj_kernel<<<dim3(64, 3), 256, 0, stream>>>(key_tokens, query_tokens,
                                               Wk, Wq, Wvd, key, qry, val);

  const size_t smem = (size_t)(NQ * QSTR + NA * WSTR + NA * LSTR + NA + NA + 256)
                      * sizeof(float);               // ~151 KB (<=320 KB/WG)
  attn_kernel<<<NB * NK, 256, smem, stream>>>(key, qry, val, Wa, vsum);

  out_kernel<<<256, 256, 0, stream>>>(vsum, Wvu, out);
}